// TextGraphModule_56796647522982
// MI455X (gfx1250) — compile-verified
//
#include <hip/hip_runtime.h>

#define NN 50000      // nodes
#define NE 800000     // edges
#define HD 128        // hidden / F_IN
#define PJ 512        // projection dim
#define NC 32         // classes
#define NG 256        // graphs
#define TE 768        // text embedding dim

typedef float v2f __attribute__((ext_vector_type(2)));
typedef float v8f __attribute__((ext_vector_type(8)));

// ---------------------------------------------------------------------------
// FP32 WMMA GEMM: C[M,N] = A[M,K] @ B[K,N] (+ bias[N] if bias != null)
// One wave per 16x16 output tile. K must be a multiple of 4; M,N of 16.
// A-frag (16x4 f32): lanes 0-15 row m hold K=k,k+1 ; lanes 16-31 hold K=k+2,k+3
// B-frag (4x16 f32): VGPR0 = rows k / k+2 striped over lanes, VGPR1 = k+1 / k+3
// C/D (16x16 f32): VGPR r -> row (half*8 + r), col = lane%16
// ---------------------------------------------------------------------------
__global__ __launch_bounds__(32) void gemm_f32_wmma(
    const float* __restrict__ A, const float* __restrict__ B,
    const float* __restrict__ bias, float* __restrict__ C,
    int M, int N, int K)
{
  const int lane = threadIdx.x;
  const int m    = lane & 15;
  const int half = lane >> 4;
  const int koff = half << 1;                 // 0 or 2
  const int tm = blockIdx.x << 4;
  const int tn = blockIdx.y << 4;

  const float* Arow = A + (size_t)(tm + m) * K + koff;
  const float* Bcol = B + (size_t)koff * N + tn + m;

  v8f acc = {};
  for (int k = 0; k < K; k += 4) {
    v2f a, b;
    a.x = Arow[k];
    a.y = Arow[k + 1];
    const float* bp = Bcol + (size_t)k * N;
    b.x = bp[0];
    b.y = bp[N];
    acc = __builtin_amdgcn_wmma_f32_16x16x4_f32(false, a, false, b,
                                                (short)0, acc, false, false);
  }
  const float bv = bias ? bias[tn + m] : 0.0f;
  float* Crow = C + (size_t)(tm + (half << 3)) * N + tn + m;
#pragma unroll
  for (int r = 0; r < 8; ++r)
    Crow[(size_t)r * N] = acc[r] + bv;
}

// ---------------------------------------------------------------------------
__global__ __launch_bounds__(256) void zero_kernel(float* __restrict__ p, int n) {
  int i = blockIdx.x * 256 + threadIdx.x;
  if (i < n) p[i] = 0.0f;
}

__global__ __launch_bounds__(256) void deg_count_kernel(
    const int* __restrict__ dst, float* __restrict__ deg, int E) {
  int e = blockIdx.x * 256 + threadIdx.x;
  if (e < E) atomicAdd(&deg[dst[e]], 1.0f);
}

__global__ __launch_bounds__(256) void dnorm_kernel(float* __restrict__ deg, int n) {
  int i = blockIdx.x * 256 + threadIdx.x;
  if (i < n) deg[i] = rsqrtf(deg[i] + 1.0f);
}

// agg[i,c] = xw[i,c]*dnorm[i]^2 + bias[c]   (self-loop contribution + bias)
__global__ __launch_bounds__(256) void self_init_kernel(
    const float* __restrict__ xw, const float* __restrict__ dnorm,
    const float* __restrict__ bias, float* __restrict__ agg, int n) {
  int i = blockIdx.x * 256 + threadIdx.x;
  if (i >= n) return;
  int node = i >> 7, c = i & 127;
  float dn = dnorm[node];
  agg[i] = xw[i] * dn * dn + bias[c];
}

// agg[dst] += xw[src] * dnorm[src]*dnorm[dst]; one wave per edge (float4/lane)
__global__ __launch_bounds__(256) void edge_scatter_kernel(
    const float* __restrict__ xw, const int* __restrict__ src,
    const int* __restrict__ dst, const float* __restrict__ dnorm,
    float* __restrict__ agg, int E) {
  unsigned tid = blockIdx.x * 256u + threadIdx.x;
  unsigned e = tid >> 5;
  if (e >= (unsigned)E) return;
  int q = (tid & 31) << 2;                    // feature group of 4
  int s = src[e], d = dst[e];
  float coef = dnorm[s] * dnorm[d];
  const float4 v = *(const float4*)(xw + (size_t)s * HD + q);
  float* out = agg + (size_t)d * HD + q;
  atomicAdd(out + 0, v.x * coef);
  atomicAdd(out + 1, v.y * coef);
  atomicAdd(out + 2, v.z * coef);
  atomicAdd(out + 3, v.w * coef);
}

__device__ __forceinline__ float lrelu(float v) { return v < 0.0f ? 0.01f * v : v; }

// per-column sum / sum-of-squares of leaky_relu(agg), 128-row chunks per block
__global__ __launch_bounds__(256) void bn_stats_kernel(
    const float* __restrict__ agg, float* __restrict__ colsum,
    float* __restrict__ colsq, int rows) {
  __shared__ float ssum[256], ssq[256];
  int c = threadIdx.x & 127;
  int half = threadIdx.x >> 7;
  int r0 = blockIdx.x * 128;
  int rend = r0 + 128; if (rend > rows) rend = rows;
  float s = 0.0f, sq = 0.0f;
  for (int r = r0 + half; r < rend; r += 2) {
    float v = lrelu(agg[(size_t)r * HD + c]);
    s += v; sq += v * v;
  }
  ssum[threadIdx.x] = s; ssq[threadIdx.x] = sq;
  __syncthreads();
  if (threadIdx.x < 128) {
    atomicAdd(&colsum[c], ssum[threadIdx.x] + ssum[threadIdx.x + 128]);
    atomicAdd(&colsq[c],  ssq[threadIdx.x]  + ssq[threadIdx.x + 128]);
  }
}

// in-place: agg = BN(leaky_relu(agg)) with affine g,b
__global__ __launch_bounds__(256) void bn_apply_kernel(
    float* __restrict__ agg, const float* __restrict__ colsum,
    const float* __restrict__ colsq, const float* __restrict__ g,
    const float* __restrict__ b, int n, float invN) {
  int i = blockIdx.x * 256 + threadIdx.x;
  if (i >= n) return;
  int c = i & 127;
  float mu = colsum[c] * invN;
  float var = colsq[c] * invN - mu * mu;
  float v = lrelu(agg[i]);
  agg[i] = (v - mu) * rsqrtf(var + 1e-5f) * g[c] + b[c];
}

__global__ __launch_bounds__(256) void pool_acc_kernel(
    const float* __restrict__ h, const int* __restrict__ batch,
    float* __restrict__ sums, float* __restrict__ cnt, int n) {
  int i = blockIdx.x * 256 + threadIdx.x;
  if (i >= n) return;
  int node = i >> 7, c = i & 127;
  int g = batch[node];
  atomicAdd(&sums[(size_t)g * HD + c], h[i]);
  if (c == 0) atomicAdd(&cnt[g], 1.0f);
}

__global__ __launch_bounds__(256) void pool_div_kernel(
    const float* __restrict__ sums, const float* __restrict__ cnt,
    float* __restrict__ gemb, int n) {
  int i = blockIdx.x * 256 + threadIdx.x;
  if (i >= n) return;
  gemb[i] = sums[i] / fmaxf(cnt[i >> 7], 1.0f);
}

__global__ __launch_bounds__(256) void gelu_kernel(
    const float* __restrict__ in, float* __restrict__ out, int n) {
  int i = blockIdx.x * 256 + threadIdx.x;
  if (i >= n) return;
  float x = in[i];
  out[i] = 0.5f * x * (1.0f + erff(x * 0.70710678118654752f));
}

// out[row] = LayerNorm(H2[row] + P1[row]) * g + be   (D = 512, block per row)
__global__ __launch_bounds__(256) void res_ln_kernel(
    const float* __restrict__ H2, const float* __restrict__ P1,
    const float* __restrict__ g, const float* __restrict__ be,
    float* __restrict__ out) {
  __shared__ float ssum[256], ssq[256];
  const int row = blockIdx.x, tid = threadIdx.x;
  float vals[2]; float s = 0.0f, sq = 0.0f;
#pragma unroll
  for (int j = 0; j < 2; ++j) {
    int c = tid + j * 256;
    float v = H2[(size_t)row * PJ + c] + P1[(size_t)row * PJ + c];
    vals[j] = v; s += v; sq += v * v;
  }
  ssum[tid] = s; ssq[tid] = sq;
  __syncthreads();
  for (int st = 128; st > 0; st >>= 1) {
    if (tid < st) { ssum[tid] += ssum[tid + st]; ssq[tid] += ssq[tid + st]; }
    __syncthreads();
  }
  float mu = ssum[0] / (float)PJ;
  float var = ssq[0] / (float)PJ - mu * mu;
  float rs = rsqrtf(var + 1e-5f);
#pragma unroll
  for (int j = 0; j < 2; ++j) {
    int c = tid + j * 256;
    out[(size_t)row * PJ + c] = (vals[j] - mu) * rs * g[c] + be[c];
  }
}

__global__ __launch_bounds__(256) void log_softmax_kernel(
    const float* __restrict__ logits, float* __restrict__ out, int rows) {
  int row = blockIdx.x * 256 + threadIdx.x;
  if (row >= rows) return;
  const float* p = logits + (size_t)row * NC;
  float mx = -INFINITY;
  for (int c = 0; c < NC; ++c) mx = fmaxf(mx, p[c]);
  float se = 0.0f;
  for (int c = 0; c < NC; ++c) se += expf(p[c] - mx);
  float lse = mx + logf(se);
  for (int c = 0; c < NC; ++c) out[(size_t)row * NC + c] = p[c] - lse;
}

// ---------------------------------------------------------------------------
static inline int cdiv(int a, int b) { return (a + b - 1) / b; }

static void run_head(const float* inp, int Din,
                     const float* W1, const float* b1,
                     const float* W2, const float* b2,
                     const float* g, const float* be,
                     const float* Wc, const float* bc,
                     float* P1, float* H1, float* H2, float* logits,
                     float* proj_out, float* logp_out, hipStream_t s) {
  gemm_f32_wmma<<<dim3(NG / 16, PJ / 16), 32, 0, s>>>(inp, W1, b1, P1, NG, PJ, Din);
  gelu_kernel<<<cdiv(NG * PJ, 256), 256, 0, s>>>(P1, H1, NG * PJ);
  gemm_f32_wmma<<<dim3(NG / 16, PJ / 16), 32, 0, s>>>(H1, W2, b2, H2, NG, PJ, PJ);
  res_ln_kernel<<<NG, 256, 0, s>>>(H2, P1, g, be, proj_out);
  gemm_f32_wmma<<<dim3(NG / 16, NC / 16), 32, 0, s>>>(proj_out, Wc, bc, logits, NG, NC, PJ);
  log_softmax_kernel<<<1, 256, 0, s>>>(logits, logp_out, NG);
}

extern "C" void kernel_launch(void* const* d_in, const int* in_sizes, int n_in,
                              void* d_out, int out_size, void* d_ws, size_t ws_size,
                              hipStream_t stream) {
  const float* x      = (const float*)d_in[0];
  const int*   ei     = (const int*)d_in[1];
  const int*   batch  = (const int*)d_in[2];
  const float* tfeat  = (const float*)d_in[3];
  const float* gcnW   = (const float*)d_in[4];
  const float* gcnB   = (const float*)d_in[5];
  const float* bnG    = (const float*)d_in[6];
  const float* bnB    = (const float*)d_in[7];
  const float* gW1 = (const float*)d_in[8],  *gb1 = (const float*)d_in[9];
  const float* gW2 = (const float*)d_in[10], *gb2 = (const float*)d_in[11];
  const float* gG  = (const float*)d_in[12], *gBe = (const float*)d_in[13];
  const float* gcW = (const float*)d_in[14], *gcB = (const float*)d_in[15];
  const float* tW1 = (const float*)d_in[16], *tb1 = (const float*)d_in[17];
  const float* tW2 = (const float*)d_in[18], *tb2 = (const float*)d_in[19];
  const float* tG  = (const float*)d_in[20], *tBe = (const float*)d_in[21];
  const float* tcW = (const float*)d_in[22], *tcB = (const float*)d_in[23];

  const int* src = ei;
  const int* dst = ei + NE;

  float* wsf = (float*)d_ws;
  size_t off = 0;
  auto alloc = [&](size_t n) { float* p = wsf + off; off += (n + 63) & ~(size_t)63; return p; };
  float* xw      = alloc((size_t)NN * HD);
  float* agg     = alloc((size_t)NN * HD);
  float* dnorm   = alloc(NN);
  float* colsum  = alloc(HD);
  float* colsq   = alloc(HD);
  float* psum    = alloc((size_t)NG * HD);
  float* pcnt    = alloc(NG);
  float* gemb    = alloc((size_t)NG * HD);
  float* P1      = alloc((size_t)NG * PJ);
  float* H1      = alloc((size_t)NG * PJ);
  float* H2      = alloc((size_t)NG * PJ);
  float* logits  = alloc((size_t)NG * NC);

  float* out_gproj = (float*)d_out;                       // [NG, PJ]
  float* out_tproj = out_gproj + (size_t)NG * PJ;         // [NG, PJ]
  float* out_glogp = out_tproj + (size_t)NG * PJ;         // [NG, NC]
  float* out_tlogp = out_glogp + (size_t)NG * NC;         // [NG, NC]

  // --- degree / symmetric normalization ---
  zero_kernel<<<cdiv(NN, 256), 256, 0, stream>>>(dnorm, NN);
  deg_count_kernel<<<cdiv(NE, 256), 256, 0, stream>>>(dst, dnorm, NE);
  dnorm_kernel<<<cdiv(NN, 256), 256, 0, stream>>>(dnorm, NN);

  // --- 4 GCN layers ---
  const float* h_in = x;
  const int nelem = NN * HD;
  for (int l = 0; l < 4; ++l) {
    const float* W = gcnW + (size_t)l * HD * HD;
    const float* b = gcnB + (size_t)l * HD;
    gemm_f32_wmma<<<dim3(NN / 16, HD / 16), 32, 0, stream>>>(
        h_in, W, nullptr, xw, NN, HD, HD);
    self_init_kernel<<<cdiv(nelem, 256), 256, 0, stream>>>(xw, dnorm, b, agg, nelem);
    edge_scatter_kernel<<<cdiv(NE * 32, 256), 256, 0, stream>>>(
        xw, src, dst, dnorm, agg, NE);
    zero_kernel<<<1, 256, 0, stream>>>(colsum, HD);
    zero_kernel<<<1, 256, 0, stream>>>(colsq, HD);
    bn_stats_kernel<<<cdiv(NN, 128), 256, 0, stream>>>(agg, colsum, colsq, NN);
    bn_apply_kernel<<<cdiv(nelem, 256), 256, 0, stream>>>(
        agg, colsum, colsq, bnG + (size_t)l * HD, bnB + (size_t)l * HD,
        nelem, 1.0f / (float)NN);
    h_in = agg;
  }

  // --- global mean pool ---
  zero_kernel<<<cdiv(NG * HD, 256), 256, 0, stream>>>(psum, NG * HD);
  zero_kernel<<<1, 256, 0, stream>>>(pcnt, NG);
  pool_acc_kernel<<<cdiv(nelem, 256), 256, 0, stream>>>(agg, batch, psum, pcnt, nelem);
  pool_div_kernel<<<cdiv(NG * HD, 256), 256, 0, stream>>>(psum, pcnt, gemb, NG * HD);

  // --- graph head, then text head (buffers reused; stream-ordered) ---
  run_head(gemb, HD, gW1, gb1, gW2, gb2, gG, gBe, gcW, gcB,
           P1, H1, H2, logits, out_gproj, out_glogp, stream);
  run_head(tfeat, TE, tW1, tb1, tW2, tb2, tG, tBe, tcW, tcB,
           P1, H1, H2, logits, out_tproj, out_tlogp, stream);
}